// CaseNet_47966194761817
// MI455X (gfx1250) — compile-verified
//
#include <hip/hip_runtime.h>
#include <stdint.h>

#define THRESH_  (-3.0f)
#define NMS_TH_  0.05f
#define KCAND    4096
#define TOPK_    300
#define NTOT     (128*128*128*3)
#define CAP      6144
#define SORTN    8192

typedef float v2f __attribute__((ext_vector_type(2)));
typedef float v8f __attribute__((ext_vector_type(8)));

// ---------- CDNA5 async global->LDS copy (ASYNCcnt path) ----------
__device__ __forceinline__ void async_ld_b128(const uint32_t* g, uint32_t* l) {
  uint64_t ga = (uint64_t)(uintptr_t)g;
  // generic pointer to __shared__ has the LDS byte offset in its low 32 bits
  uint32_t la = (uint32_t)(uintptr_t)l;
  asm volatile("global_load_async_to_lds_b128 %0, %1, off"
               :: "v"(la), "v"(ga) : "memory");
}
__device__ __forceinline__ void wait_async_le8()  { asm volatile("s_wait_asynccnt 0x8" ::: "memory"); }
__device__ __forceinline__ void wait_async_le0()  { asm volatile("s_wait_asynccnt 0x0" ::: "memory"); }

// order-preserving float->uint key; 0 marks invalid (score <= THRESH or NaN)
__device__ __forceinline__ uint32_t score_key(float s) {
  if (!(s > THRESH_)) return 0u;
  uint32_t u = __float_as_uint(s);
  return u ^ ((u & 0x80000000u) ? 0xFFFFFFFFu : 0x80000000u);
}

// ---------------- kernels ----------------
__global__ void k_zero(uint32_t* p, int n) {
  int i = blockIdx.x * 256 + threadIdx.x;
  if (i < n) p[i] = 0u;
}

__global__ __launch_bounds__(256) void k_keys_hist1(const float* __restrict__ in,
                                                    uint32_t* __restrict__ keys,
                                                    uint32_t* __restrict__ hist1) {
  __shared__ uint32_t lh[4096];
  for (int b = threadIdx.x; b < 4096; b += 256) lh[b] = 0u;
  __syncthreads();
  int base = blockIdx.x * 2048 + threadIdx.x;
#pragma unroll
  for (int r = 0; r < 8; ++r) {
    int i = base + r * 256;
    float s = in[(size_t)i * 5];
    uint32_t k = score_key(s);
    keys[i] = k;
    if (k) atomicAdd(&lh[k >> 20], 1u);
  }
  __syncthreads();
  for (int b = threadIdx.x; b < 4096; b += 256) {
    uint32_t v = lh[b];
    if (v) atomicAdd(&hist1[b], v);
  }
}

// ---- WMMA-based parallel suffix scan over a 4096-bin histogram ----
// S16 = Tri(16x16, Tri[r][k] = k>=r) x X(16x16) computes 16 independent
// 16-point inclusive suffix scans (one per column). K=16 is accumulated as
// 4 chained V_WMMA_F32_16X16X4_F32 issues.
__device__ __forceinline__ v8f tri_mul_acc(const float* Xlds, v8f acc, int lane) {
  int half = lane >> 4;   // lane half selects K+2 per ISA A/B layout
  int n    = lane & 15;   // A: output row; B: column
#pragma unroll
  for (int k4 = 0; k4 < 4; ++k4) {
    int k0 = 4 * k4 + 2 * half;
    v2f a, b;
    a.x = (k0     >= n) ? 1.0f : 0.0f;   // Tri[row=n][k0]
    a.y = (k0 + 1 >= n) ? 1.0f : 0.0f;   // Tri[row=n][k0+1]
    b.x = Xlds[n * 16 + k0];             // X[k0][col=n]   (col-major tile)
    b.y = Xlds[n * 16 + k0 + 1];
    acc = __builtin_amdgcn_wmma_f32_16x16x4_f32(false, a, false, b,
                                                (short)0, acc, false, false);
  }
  return acc;
}

__device__ __forceinline__ void store_cd(float* out, v8f d, int lane) {
  int col = lane & 15;
  int rb  = (lane >> 4) * 8;             // C/D layout: VGPR v -> row v (+8 for hi half)
#pragma unroll
  for (int v = 0; v < 8; ++v) out[col * 16 + rb + v] = d[v];
}

// mode 0: hist1 -> ctr[0]=bin, ctr[1]=count strictly above
// mode 1: hist2 -> ctr[2]=24-bit threshold
__global__ __launch_bounds__(32) void k_scan_wmma(const uint32_t* __restrict__ hist,
                                                  uint32_t* __restrict__ ctr, int mode) {
  __shared__ float h[4096];
  __shared__ float sfx1[4096];
  __shared__ float g[256];
  __shared__ float sfx2[256];
  __shared__ float sfx3[16];
  __shared__ uint32_t redI[32];
  __shared__ float    redV[32];
  int lane = threadIdx.x;
  for (int i = lane; i < 4096; i += 32) h[i] = (float)hist[i];
  __syncthreads();
  // level 1: 16 tiles of 256 bins, each = one Tri x X product (4 wmma)
  for (int p = 0; p < 16; ++p) {
    v8f acc; 
#pragma unroll
    for (int q = 0; q < 8; ++q) acc[q] = 0.0f;
    acc = tri_mul_acc(&h[256 * p], acc, lane);
    store_cd(&sfx1[256 * p], acc, lane);
    __syncthreads();
  }
  for (int j = lane; j < 256; j += 32) g[j] = sfx1[j * 16];  // tile totals
  __syncthreads();
  // level 2: suffix scan of 256 tile totals
  {
    v8f acc;
#pragma unroll
    for (int q = 0; q < 8; ++q) acc[q] = 0.0f;
    acc = tri_mul_acc(g, acc, lane);
    store_cd(sfx2, acc, lane);
  }
  __syncthreads();
  // level 3: 16 group totals (tiny, serial)
  if (lane == 0) {
    float run = 0.0f;
    for (int m = 15; m >= 0; --m) { run += sfx2[m * 16]; sfx3[m] = run; }
  }
  __syncthreads();
  float target = (mode == 0) ? (float)KCAND : (float)(KCAND - (int)ctr[1]);
  // find largest b with global inclusive suffix >= target
  int best = -1; float bestS = 0.0f;
  for (int b = lane; b < 4096; b += 32) {
    int t = b >> 4, grp = b >> 8;
    float S = sfx1[b] + (sfx2[t] - g[t]) + (sfx3[grp] - sfx2[grp * 16]);
    if (S >= target) { best = b; bestS = S; }   // ascending loop -> keeps max b
  }
  redI[lane] = (uint32_t)(best + 1);
  redV[lane] = bestS;
  __syncthreads();
  if (lane == 0) {
    int bb = 0; float bs = 0.0f;
    for (int q = 0; q < 32; ++q)
      if ((int)redI[q] > bb) { bb = (int)redI[q]; bs = redV[q]; }
    if (mode == 0) {
      if (bb > 0) { int b = bb - 1; ctr[0] = (uint32_t)b; ctr[1] = (uint32_t)(bs - h[b]); }
      else        { ctr[0] = 0u; ctr[1] = 0u; }   // pathological: < KCAND valid
    } else {
      uint32_t B1 = ctr[0];
      ctr[2] = (bb > 0) ? ((B1 << 12) | (uint32_t)(bb - 1)) : (B1 << 12);
    }
  }
}

__global__ __launch_bounds__(256) void k_hist2(const uint32_t* __restrict__ keys,
                                               const uint32_t* __restrict__ ctr,
                                               uint32_t* __restrict__ hist2) {
  __shared__ uint32_t lh[4096];
  for (int b = threadIdx.x; b < 4096; b += 256) lh[b] = 0u;
  __syncthreads();
  uint32_t B1 = ctr[0];
  int base = blockIdx.x * 2048 + threadIdx.x;
#pragma unroll
  for (int r = 0; r < 8; ++r) {
    uint32_t k = keys[base + r * 256];
    if (k && (k >> 20) == B1) atomicAdd(&lh[(k >> 8) & 0xFFFu], 1u);
  }
  __syncthreads();
  for (int b = threadIdx.x; b < 4096; b += 256) {
    uint32_t v = lh[b];
    if (v) atomicAdd(&hist2[b], v);
  }
}

__global__ __launch_bounds__(256) void k_collect(const uint32_t* __restrict__ keys,
                                                 uint32_t* __restrict__ ctr,
                                                 uint2* __restrict__ candkv) {
  uint32_t thr = ctr[2];
  int base = blockIdx.x * 2048 + threadIdx.x;
#pragma unroll
  for (int r = 0; r < 8; ++r) {
    int i = base + r * 256;
    uint32_t k = keys[i];
    if (k && (k >> 8) >= thr) {
      uint32_t pos = atomicAdd(&ctr[3], 1u);
      if (pos < CAP) candkv[pos] = make_uint2(k, (uint32_t)i);
    }
  }
}

// single-WG bitonic sort (desc by key, asc by index) + box decode; 64KB LDS
__global__ __launch_bounds__(1024) void k_sort_decode(const float* __restrict__ in,
                                                      const uint2* __restrict__ candkv,
                                                      const uint32_t* __restrict__ ctr,
                                                      float* __restrict__ cand,
                                                      uint32_t* __restrict__ validbits) {
  __shared__ uint64_t arr[SORTN];
  int tid = threadIdx.x;
  int C = (int)ctr[3]; if (C > CAP) C = CAP;
  for (int i = tid; i < SORTN; i += 1024) {
    uint64_t v = 0ull;
    if (i < C) {
      uint2 kv = candkv[i];
      v = ((uint64_t)kv.x << 32) | (uint64_t)(uint32_t)(~kv.y);
    }
    arr[i] = v;
  }
  __syncthreads();
  for (int k = 2; k <= SORTN; k <<= 1) {
    for (int j = k >> 1; j > 0; j >>= 1) {
      for (int i = tid; i < SORTN; i += 1024) {
        int ixj = i ^ j;
        if (ixj > i) {
          uint64_t a = arr[i], b = arr[ixj];
          bool up = ((i & k) == 0);
          if (up ? (a < b) : (a > b)) { arr[i] = b; arr[ixj] = a; }  // descending
        }
      }
      __syncthreads();
    }
  }
  const float anch[3] = {10.f, 30.f, 60.f};
  for (int t = tid; t < KCAND; t += 1024) {
    uint64_t v = arr[t];
    uint32_t idx = (v != 0ull) ? (uint32_t)(~((uint32_t)v)) : (uint32_t)t;
    size_t off = (size_t)idx * 5;
    float s  = in[off + 0];
    float dz = in[off + 1], dy = in[off + 2], dx = in[off + 3], dd = in[off + 4];
    uint32_t a = idx % 3u;  uint32_t r0 = idx / 3u;
    uint32_t wx = r0 & 127u; uint32_t r1 = r0 >> 7;
    uint32_t hy = r1 & 127u; uint32_t zz = r1 >> 7;
    float an = anch[a];
    cand[t * 5 + 0] = s;
    cand[t * 5 + 1] = 1.5f + 4.f * (float)zz + dz * an;
    cand[t * 5 + 2] = 1.5f + 4.f * (float)hy + dy * an;
    cand[t * 5 + 3] = 1.5f + 4.f * (float)wx + dx * an;
    cand[t * 5 + 4] = expf(dd) * an;
  }
  __syncthreads();
  if (tid < 128) {
    uint32_t wbits = 0;
    for (int c = 0; c < 32; ++c)
      if (arr[tid * 32 + c] != 0ull) wbits |= (1u << c);
    validbits[tid] = wbits;
  }
}

// 4096x4096 suppression bitmask; one row per block, cols staged in LDS
__global__ __launch_bounds__(128) void k_mask(const float* __restrict__ cand,
                                              uint32_t* __restrict__ mask) {
  __shared__ float4 cb[KCAND];   // (z,y,x,d)
  int tid = threadIdx.x;
  for (int j = tid; j < KCAND; j += 128)
    cb[j] = make_float4(cand[j*5+1], cand[j*5+2], cand[j*5+3], cand[j*5+4]);
  __syncthreads();
  int i = blockIdx.x;
  float4 r = cb[i];
  float rr   = r.w * 0.5f;
  float rvol = r.w * r.w * r.w;
  uint32_t word = 0;
  int jbase = tid * 32;
#pragma unroll 4
  for (int c = 0; c < 32; ++c) {
    float4 b = cb[jbase + c];
    float br = b.w * 0.5f;
    float oz = fmaxf(fminf(r.x + rr, b.x + br) - fmaxf(r.x - rr, b.x - br), 0.f);
    float oy = fmaxf(fminf(r.y + rr, b.y + br) - fmaxf(r.y - rr, b.y - br), 0.f);
    float ox = fmaxf(fminf(r.z + rr, b.z + br) - fmaxf(r.z - rr, b.z - br), 0.f);
    float inter = oz * oy * ox;
    float bvol  = b.w * b.w * b.w;
    float iou   = inter / (rvol + bvol - inter);
    if (iou >= NMS_TH_) word |= (1u << c);
  }
  mask[(size_t)i * 128 + tid] = word;
}

// sequential greedy NMS, 32-candidate blocks; next block's 16KB of mask rows
// double-buffered into LDS via CDNA5 async copies (ASYNCcnt)
__global__ __launch_bounds__(128) void k_nms(const uint32_t* __restrict__ mask,
                                             const uint32_t* __restrict__ validbits,
                                             uint32_t* __restrict__ keepout) {
  __shared__ __align__(16) uint32_t rowbuf[2][4096];  // 2 x 16KB tiles
  __shared__ uint32_t keepw[128];
  __shared__ uint32_t keptword;
  int tid = threadIdx.x;
  keepw[tid] = validbits[tid];
  {
    const uint32_t* g = mask;            // tile 0
#pragma unroll
    for (int r = 0; r < 8; ++r) {
      int o = r * 512 + tid * 4;
      async_ld_b128(g + o, &rowbuf[0][o]);
    }
  }
  int cur = 0;
  for (int b = 0; b < 128; ++b) {
    if (b + 1 < 128) {
      const uint32_t* g = mask + (size_t)(b + 1) * 4096;
      int nxt = cur ^ 1;
#pragma unroll
      for (int r = 0; r < 8; ++r) {
        int o = r * 512 + tid * 4;
        async_ld_b128(g + o, &rowbuf[nxt][o]);
      }
      wait_async_le8();   // current tile's 8 per-wave copies are done (in-order)
    } else {
      wait_async_le0();
    }
    __syncthreads();
    if (tid < 32) {       // wave 0: intra-block greedy resolve at register speed
      uint32_t m = rowbuf[cur][tid * 128 + b];
      uint32_t k = keepw[b];
#pragma unroll 1
      for (int i = 0; i < 32; ++i) {
        if ((k >> i) & 1u) {
          uint32_t mi = __builtin_amdgcn_readlane(m, i);
          k &= ~(mi & (0xFFFFFFFEu << i));    // suppress only j > i
        }
      }
      if (tid == 0) { keepw[b] = k; keptword = k; }
    }
    __syncthreads();
    if (tid > b) {        // parallel update of all later keep-words
      uint32_t K = keptword, orv = 0;
      while (K) {
        int i = __ffs(K) - 1; K &= K - 1;
        orv |= rowbuf[cur][i * 128 + tid];
      }
      keepw[tid] &= ~orv;
    }
    __syncthreads();
    cur ^= 1;
  }
  keepout[tid] = keepw[tid];
}

__global__ __launch_bounds__(128) void k_output(const float* __restrict__ cand,
                                                const uint32_t* __restrict__ keepin,
                                                float* __restrict__ out) {
  __shared__ uint32_t pre[129];
  int tid = threadIdx.x;
  uint32_t w = keepin[tid];
  pre[tid] = (uint32_t)__popc(w);
  __syncthreads();
  if (tid == 0) {
    uint32_t run = 0;
    for (int t = 0; t < 128; ++t) { uint32_t c = pre[t]; pre[t] = run; run += c; }
    pre[128] = run;
  }
  __syncthreads();
  uint32_t keptTotal = pre[128];
  uint32_t base = pre[tid];
  uint32_t cnt = 0;
  for (int c = 0; c < 32; ++c) {
    int j = tid * 32 + c;
    if ((w >> c) & 1u) {
      uint32_t slot = base + cnt; ++cnt;
      if (slot < TOPK_)
        for (int q = 0; q < 5; ++q) out[slot * 5 + q] = cand[j * 5 + q];
    }
  }
  if (keptTotal < TOPK_) {  // pad with suppressed candidates, index-ascending
    uint32_t nkbase = (uint32_t)(tid * 32) - base;
    uint32_t cnt2 = 0;
    for (int c = 0; c < 32; ++c) {
      int j = tid * 32 + c;
      if (!((w >> c) & 1u)) {
        uint32_t slot = keptTotal + nkbase + cnt2; ++cnt2;
        if (slot < TOPK_)
          for (int q = 0; q < 5; ++q) out[slot * 5 + q] = cand[j * 5 + q];
      }
    }
  }
}

extern "C" void kernel_launch(void* const* d_in, const int* in_sizes, int n_in,
                              void* d_out, int out_size, void* d_ws, size_t ws_size,
                              hipStream_t stream) {
  (void)in_sizes; (void)n_in; (void)out_size; (void)ws_size;
  const float* in = (const float*)d_in[0];
  float* out = (float*)d_out;

  uint8_t* w = (uint8_t*)d_ws;
  size_t o = 0;
  auto take = [&](size_t bytes) -> size_t {
    size_t r = (o + 255) & ~(size_t)255; o = r + bytes; return r;
  };
  uint32_t* keys   = (uint32_t*)(w + take((size_t)NTOT * 4));
  uint32_t* meta   = (uint32_t*)(w + take((4096 + 4096 + 16) * 4));
  uint32_t* hist1  = meta;
  uint32_t* hist2  = meta + 4096;
  uint32_t* ctr    = meta + 8192;
  uint2*    candkv = (uint2*)   (w + take((size_t)CAP * 8));
  float*    cand   = (float*)   (w + take((size_t)KCAND * 5 * 4));
  uint32_t* validb = (uint32_t*)(w + take(128 * 4));
  uint32_t* keepb  = (uint32_t*)(w + take(128 * 4));
  uint32_t* mask   = (uint32_t*)(w + take((size_t)KCAND * 128 * 4));

  k_zero<<<(8208 + 255) / 256, 256, 0, stream>>>(meta, 8208);
  k_keys_hist1<<<3072, 256, 0, stream>>>(in, keys, hist1);
  k_scan_wmma<<<1, 32, 0, stream>>>(hist1, ctr, 0);
  k_hist2<<<3072, 256, 0, stream>>>(keys, ctr, hist2);
  k_scan_wmma<<<1, 32, 0, stream>>>(hist2, ctr, 1);
  k_collect<<<3072, 256, 0, stream>>>(keys, ctr, candkv);
  k_sort_decode<<<1, 1024, 0, stream>>>(in, candkv, ctr, cand, validb);
  k_mask<<<KCAND, 128, 0, stream>>>(cand, mask);
  k_nms<<<1, 128, 0, stream>>>(mask, validb, keepb);
  k_output<<<1, 128, 0, stream>>>(cand, keepb, out);
}